// CPSAMONNX_46033459479210
// MI455X (gfx1250) — compile-verified
//
#include <hip/hip_runtime.h>
#include <cmath>

// ---------------------------------------------------------------------------
// Pass 1: pack the two planar channels of dP into an interleaved float2 field
// with the per-channel scale (2 / (size-1)) pre-applied.
//   pk[y*W+x].x = dP[1][y][x] * 2/(H-1)   (added to pt.x each step)
//   pk[y*W+x].y = dP[0][y][x] * 2/(W-1)   (added to pt.y each step)
// Halves the gather count in the hot loop and makes the 2x2 bilinear footprint
// cacheline-local. The 33.5MB packed field is fully L2-resident (192MB L2).
// ---------------------------------------------------------------------------
__global__ void cps_pack_kernel(const float* __restrict__ dP,
                                float2* __restrict__ pk,
                                int HW, float sx, float sy) {
    int i = blockIdx.x * blockDim.x + threadIdx.x;
    if (i < HW) {
        float a = dP[HW + i];   // channel 1 -> x-increment
        float b = dP[i];        // channel 0 -> y-increment
        pk[i] = make_float2(a * sx, b * sy);
    }
}

// one bilinear step for a single point, packed field
__device__ __forceinline__ void cps_step(const float2* __restrict__ pk,
                                         int W, int H, float fW, float fH,
                                         float& px, float& py) {
    // grid_sample coords, align_corners=False
    float fx = ((px + 1.0f) * fW - 1.0f) * 0.5f;
    float fy = ((py + 1.0f) * fH - 1.0f) * 0.5f;
    float x0f = floorf(fx), y0f = floorf(fy);
    float wx1 = fx - x0f,  wy1 = fy - y0f;
    float wx0 = 1.0f - wx1, wy0 = 1.0f - wy1;
    int ix0 = (int)x0f, iy0 = (int)y0f;
    int ix1 = ix0 + 1,  iy1 = iy0 + 1;

    // clamp for a valid address, zero the weight if the tap is out of bounds
    int cx0 = max(ix0, 0), cx1 = min(ix1, W - 1);
    int cy0 = max(iy0, 0), cy1 = min(iy1, H - 1);
    float ax0 = ((unsigned)ix0 < (unsigned)W) ? wx0 : 0.0f;
    float ax1 = ((unsigned)ix1 < (unsigned)W) ? wx1 : 0.0f;
    float ay0 = ((unsigned)iy0 < (unsigned)H) ? wy0 : 0.0f;
    float ay1 = ((unsigned)iy1 < (unsigned)H) ? wy1 : 0.0f;

    const float2* __restrict__ r0 = pk + (size_t)cy0 * (size_t)W;
    const float2* __restrict__ r1 = pk + (size_t)cy1 * (size_t)W;
    float2 t00 = r0[cx0];
    float2 t01 = r0[cx1];
    float2 t10 = r1[cx0];
    float2 t11 = r1[cx1];

    float w00 = ax0 * ay0, w01 = ax1 * ay0;
    float w10 = ax0 * ay1, w11 = ax1 * ay1;
    float dx = t00.x * w00 + t01.x * w01 + t10.x * w10 + t11.x * w11;
    float dy = t00.y * w00 + t01.y * w01 + t10.y * w10 + t11.y * w11;

    px = fminf(fmaxf(px + dx, -1.0f), 1.0f);
    py = fminf(fmaxf(py + dy, -1.0f), 1.0f);
}

// ---------------------------------------------------------------------------
// Pass 2 (packed path): TWO points per thread -> two independent dependency
// chains per lane -> 8 outstanding global_load_b64 per iteration instead of 4,
// and each chain's VALU work overlaps the other's load latency. 4096 waves
// still saturates every SIMD slot on the device.
// ---------------------------------------------------------------------------
__global__ void cps_track_packed2(const float2* __restrict__ pk,
                                  const int* __restrict__ inds,
                                  const int* __restrict__ niter_p,
                                  float* __restrict__ out,
                                  int N, int W, int H) {
    const int half = (N + 1) / 2;
    int t = blockIdx.x * blockDim.x + threadIdx.x;
    if (t >= half) return;

    const int niter = niter_p[0];
    const float fW = (float)W, fH = (float)H;
    const float inv0 = (float)(H - 1);   // sizes[0]
    const float inv1 = (float)(W - 1);   // sizes[1]

    const int iA = t;
    const int iBreal = t + half;
    const bool hasB = iBreal < N;
    const int iB = hasB ? iBreal : iA;   // duplicate chain A's point if N odd

    // pt0 = inds_swapped / sizes * 2 - 1
    float pxA = (float)inds[N + iA] / inv0 * 2.0f - 1.0f;
    float pyA = (float)inds[iA]     / inv1 * 2.0f - 1.0f;
    float pxB = (float)inds[N + iB] / inv0 * 2.0f - 1.0f;
    float pyB = (float)inds[iB]     / inv1 * 2.0f - 1.0f;

    for (int it = 0; it < niter; ++it) {
        cps_step(pk, W, H, fW, fH, pxA, pyA);
        cps_step(pk, W, H, fW, fH, pxB, pyB);
    }

    // out row0 = y in pixels (scaled by W-1), row1 = x (scaled by H-1)
    out[iA]     = (pyA + 1.0f) * 0.5f * inv1;
    out[N + iA] = (pxA + 1.0f) * 0.5f * inv0;
    if (hasB) {
        out[iB]     = (pyB + 1.0f) * 0.5f * inv1;
        out[N + iB] = (pxB + 1.0f) * 0.5f * inv0;
    }
}

// ---------------------------------------------------------------------------
// Fallback (workspace too small): read the planar field directly (8 gathers),
// apply channel scales after sampling. Identical math otherwise.
// ---------------------------------------------------------------------------
__global__ void cps_track_planar(const float* __restrict__ dP,
                                 const int* __restrict__ inds,
                                 const int* __restrict__ niter_p,
                                 float* __restrict__ out,
                                 int N, int W, int H) {
    int i = blockIdx.x * blockDim.x + threadIdx.x;
    if (i >= N) return;

    const int niter = niter_p[0];
    const int HW = H * W;
    const float fW = (float)W, fH = (float)H;
    const float sx = 2.0f / (float)(H - 1);
    const float sy = 2.0f / (float)(W - 1);

    float px = (float)inds[N + i] / (float)(H - 1) * 2.0f - 1.0f;
    float py = (float)inds[i]     / (float)(W - 1) * 2.0f - 1.0f;

    for (int it = 0; it < niter; ++it) {
        float fx = ((px + 1.0f) * fW - 1.0f) * 0.5f;
        float fy = ((py + 1.0f) * fH - 1.0f) * 0.5f;
        float x0f = floorf(fx), y0f = floorf(fy);
        float wx1 = fx - x0f,  wy1 = fy - y0f;
        float wx0 = 1.0f - wx1, wy0 = 1.0f - wy1;
        int ix0 = (int)x0f, iy0 = (int)y0f;
        int ix1 = ix0 + 1,  iy1 = iy0 + 1;

        int cx0 = max(ix0, 0), cx1 = min(ix1, W - 1);
        int cy0 = max(iy0, 0), cy1 = min(iy1, H - 1);
        float ax0 = ((unsigned)ix0 < (unsigned)W) ? wx0 : 0.0f;
        float ax1 = ((unsigned)ix1 < (unsigned)W) ? wx1 : 0.0f;
        float ay0 = ((unsigned)iy0 < (unsigned)H) ? wy0 : 0.0f;
        float ay1 = ((unsigned)iy1 < (unsigned)H) ? wy1 : 0.0f;

        size_t o00 = (size_t)cy0 * W + cx0;
        size_t o01 = (size_t)cy0 * W + cx1;
        size_t o10 = (size_t)cy1 * W + cx0;
        size_t o11 = (size_t)cy1 * W + cx1;

        float w00 = ax0 * ay0, w01 = ax1 * ay0;
        float w10 = ax0 * ay1, w11 = ax1 * ay1;

        float dx = (dP[HW + o00] * w00 + dP[HW + o01] * w01 +
                    dP[HW + o10] * w10 + dP[HW + o11] * w11) * sx;
        float dy = (dP[o00] * w00 + dP[o01] * w01 +
                    dP[o10] * w10 + dP[o11] * w11) * sy;

        px = fminf(fmaxf(px + dx, -1.0f), 1.0f);
        py = fminf(fmaxf(py + dy, -1.0f), 1.0f);
    }

    out[i]     = (py + 1.0f) * 0.5f * (float)(W - 1);
    out[N + i] = (px + 1.0f) * 0.5f * (float)(H - 1);
}

extern "C" void kernel_launch(void* const* d_in, const int* in_sizes, int n_in,
                              void* d_out, int out_size, void* d_ws, size_t ws_size,
                              hipStream_t stream) {
    const float* dP     = (const float*)d_in[0];   // (2,H,W) fp32
    const int*   inds   = (const int*)d_in[1];     // (2,N) int32
    const int*   niter  = (const int*)d_in[2];     // scalar on device
    float*       out    = (float*)d_out;           // (2,N) fp32

    const int HW = in_sizes[0] / 2;                // H*W
    const int W  = (int)llround(sqrt((double)HW)); // square field (2048)
    const int H  = HW / W;
    const int N  = in_sizes[1] / 2;

    const int BS = 256;
    const size_t need = (size_t)HW * sizeof(float2);

    if (ws_size >= need) {
        float2* pk = (float2*)d_ws;
        cps_pack_kernel<<<(HW + BS - 1) / BS, BS, 0, stream>>>(
            dP, pk, HW, 2.0f / (float)(H - 1), 2.0f / (float)(W - 1));
        const int half = (N + 1) / 2;
        cps_track_packed2<<<(half + BS - 1) / BS, BS, 0, stream>>>(
            pk, inds, niter, out, N, W, H);
    } else {
        cps_track_planar<<<(N + BS - 1) / BS, BS, 0, stream>>>(
            dP, inds, niter, out, N, W, H);
    }
}